// DiracAuditor_32530082300275
// MI455X (gfx1250) — compile-verified
//
#include <hip/hip_runtime.h>
#include <math.h>

typedef __attribute__((ext_vector_type(2))) float v2f;
typedef __attribute__((ext_vector_type(8))) float v8f;

namespace {
// Constants mirrored from the reference (computed in f64, cast to f32).
constexpr double SQ3  = 1.7320508075688772935274463415059;
constexpr double PI_D = 3.1415926535897932384626433832795;
constexpr double ACCD = 2.46 / SQ3;                      // A_CC
constexpr float  THOP = 2.8f;
constexpr float  D0X  = 0.0f;
constexpr float  D0Y  = (float)(ACCD * -1.0);
constexpr float  D1X  = (float)(ACCD * (SQ3 * 0.5));     // = 1.23
constexpr float  D1Y  = (float)(ACCD * 0.5);
constexpr float  D2X  = (float)(-(ACCD * (SQ3 * 0.5)));
constexpr float  D2Y  = (float)(ACCD * 0.5);
constexpr float  KXA  = (float)(4.0 * PI_D / (3.0 * SQ3 * ACCD));
constexpr float  BERRY_R   = (float)(0.02 / 2.46);
constexpr int    DIRS      = 2048;
constexpr int    NEWTON_IT = 100;
constexpr float  DET_TOL   = 1e-12f;
constexpr float  OFF_SCALE = 0.001f;
constexpr float  PI_F      = 3.14159265358979323846f;
constexpr float  TWO_PI_F  = 6.28318530717958647692f;
}

__device__ __forceinline__ float wrapf(float x) {
    // jnp.mod(x + pi, 2pi) - pi
    float y = x + PI_F;
    y = y - TWO_PI_F * floorf(y / TWO_PI_F);
    return y - PI_F;
}

__device__ __forceinline__ void phases(float kx, float ky,
                                       float& c0, float& s0,
                                       float& c1, float& s1,
                                       float& c2, float& s2) {
    float p0 = D0X * kx + D0Y * ky;
    float p1 = D1X * kx + D1Y * ky;
    float p2 = D2X * kx + D2Y * ky;
    sincosf(p0, &s0, &c0);
    sincosf(p1, &s1, &c1);
    sincosf(p2, &s2, &c2);
}

__global__ __launch_bounds__(256) void
DiracAuditor_berry_kernel(const float* __restrict__ off, float* __restrict__ out) {
    __shared__ float sU[DIRS][2];   // unit phasor u_j = -conj(f)/|f| per ring point
    __shared__ float sK[2];
    __shared__ float sWave[8];

    const int tid = threadIdx.x;
    const int b   = blockIdx.x;

    // ---- Damped Newton refine (sequential -> single thread per block) ----
    if (tid == 0) {
        float kx = KXA + OFF_SCALE * off[2 * b + 0];
        float ky = 0.0f + OFF_SCALE * off[2 * b + 1];
        float alpha = 1.0f;
        #pragma unroll 1
        for (int it = 0; it < NEWTON_IT; ++it) {
            float c0, s0, c1, s1, c2, s2;
            phases(kx, ky, c0, s0, c1, s1, c2, s2);
            float fr = -THOP * (c0 + c1 + c2);
            float fi = -THOP * (s0 + s1 + s2);
            // g = -t * sum_d i*delta_d*e^{i p_d}
            float gxr =  THOP * (D0X * s0 + D1X * s1 + D2X * s2);
            float gxi = -THOP * (D0X * c0 + D1X * c1 + D2X * c2);
            float gyr =  THOP * (D0Y * s0 + D1Y * s1 + D2Y * s2);
            float gyi = -THOP * (D0Y * c0 + D1Y * c1 + D2Y * c2);
            // J = [[Re gx, Re gy],[Im gx, Im gy]], F = [Re f, Im f]
            float det = gxr * gyi - gyr * gxi;
            bool  ok  = fabsf(det) >= DET_TOL;
            float sd  = ok ? det : 1.0f;
            float dkx = ( gyi * fr - gyr * fi) / sd;
            float dky = (-gxi * fr + gxr * fi) / sd;
            if (!ok) { dkx = 0.0f; dky = 0.0f; }
            float na = ok ? fminf(alpha * 2.0f, 1.0f) : alpha * 0.5f;
            kx -= na * dkx;
            ky -= na * dky;
            alpha = na;
        }
        sK[0] = kx; sK[1] = ky;
    }
    __syncthreads();
    const float Kx = sK[0], Ky = sK[1];

    // ---- Ring phasors staged in LDS: u = -conj(f)/|f| ----
    const float step = TWO_PI_F / (float)DIRS;
    for (int j = tid; j < DIRS; j += 256) {
        float th = (float)j * step;
        float st, ct;
        sincosf(th, &st, &ct);
        float kx = Kx + BERRY_R * ct;
        float ky = Ky + BERRY_R * st;
        float c0, s0, c1, s1, c2, s2;
        phases(kx, ky, c0, s0, c1, s1, c2, s2);
        float fr = -THOP * (c0 + c1 + c2);
        float fi = -THOP * (s0 + s1 + s2);
        float m  = sqrtf(fr * fr + fi * fi);
        sU[j][0] = -fr / m;   // Re u
        sU[j][1] =  fi / m;   // Im u
    }
    __syncthreads();

    // ---- Wrapped overlap phases around the closed ring ----
    float partial = 0.0f;
    for (int j = tid; j < DIRS; j += 256) {
        int   jn = (j + 1) & (DIRS - 1);
        float ar = sU[j][0],  ai = sU[j][1];
        float br = sU[jn][0], bi = sU[jn][1];
        // ov = 0.5*(1 + conj(u_j)*u_{j+1})
        float ovr = 0.5f * (1.0f + ar * br + ai * bi);
        float ovi = 0.5f * (ar * bi - ai * br);
        partial += wrapf(atan2f(ovi, ovr));
    }

    // ---- Wave reduction via V_WMMA_F32_16X16X4_F32 (exact f32 adds) ----
    // A (16x4): lane L<16 -> A[L][0]=partial; lane L>=16 -> A[L-16][2]=partial.
    // B = ones(4x16)  =>  D[m][n] = partial_m + partial_{m+16} (all n equal).
    // d[0..7] in lanes 0-15 hold rows 0-7; in lanes 16-31 rows 8-15.
    v2f a;    a.x = partial; a.y = 0.0f;
    v2f ones; ones.x = 1.0f; ones.y = 1.0f;
    v8f c = {};
    v8f d = __builtin_amdgcn_wmma_f32_16x16x4_f32(
        false, a, false, ones, (short)0, c, false, false);
    float s = d[0] + d[1] + d[2] + d[3] + d[4] + d[5] + d[6] + d[7];
    float waveTot = s + __shfl_xor(s, 16, 32);   // low half + high half

    const int wave = tid >> 5;
    const int lane = tid & 31;
    if (lane == 0) sWave[wave] = waveTot;
    __syncthreads();
    if (tid == 0) {
        float r = 0.0f;
        #pragma unroll
        for (int w = 0; w < 8; ++w) r += sWave[w];
        out[b] = r;
    }
}

extern "C" void kernel_launch(void* const* d_in, const int* in_sizes, int n_in,
                              void* d_out, int out_size, void* d_ws, size_t ws_size,
                              hipStream_t stream) {
    (void)n_in; (void)d_ws; (void)ws_size;
    const float* off = (const float*)d_in[0];
    float*       out = (float*)d_out;
    int nloops = in_sizes[0] / 2;       // [N_LOOPS, 2] flattened
    if (nloops <= 0) nloops = out_size; // defensive
    DiracAuditor_berry_kernel<<<nloops, 256, 0, stream>>>(off, out);
}